// MitoGraphEncoder_90228672955154
// MI455X (gfx1250) — compile-verified
//
#include <hip/hip_runtime.h>
#include <hip/hip_bf16.h>
#include <math.h>

typedef __attribute__((ext_vector_type(2))) float v2f;
typedef __attribute__((ext_vector_type(8))) float v8f;

#define TPB 256

// ---------------- utility kernels ----------------

__global__ void fill_kernel(float* __restrict__ p, float v, int n) {
  int i = blockIdx.x * blockDim.x + threadIdx.x;
  if (i < n) p[i] = v;
}

// h[n,64] = x[n,IN] @ w[IN,64] + b   (tiny IN: 3..64, scalar is fine)
__global__ void proj_kernel(const float* __restrict__ x, const float* __restrict__ w,
                            const float* __restrict__ b, float* __restrict__ h,
                            int N, int IN) {
  int idx = blockIdx.x * blockDim.x + threadIdx.x;
  if (idx >= N * 64) return;
  int n = idx >> 6, o = idx & 63;
  float s = b[o];
  const float* xr = x + (size_t)n * IN;
  for (int i = 0; i < IN; ++i) s = fmaf(xr[i], w[i * 64 + o], s);
  h[idx] = s;
}

// Y[N,OUTC] = X[N,64] @ W[64,OUTC] + bias, f32 WMMA 16x16x4.
// Block = 8 waves x 16 rows. W staged in LDS; A fragments clamped (no divergent
// guards: garbage rows only produce garbage D rows, which are never stored).
template <int OUTC>
__global__ __launch_bounds__(256)
void gemm_bias_wmma(const float* __restrict__ X, const float* __restrict__ W,
                    const float* __restrict__ bias, float* __restrict__ Y, int N) {
  __shared__ float Wlds[64 * OUTC];
  {
    constexpr int TOT4 = 64 * OUTC / 4;
    const float4* Wv = (const float4*)W;
    float4* Lv = (float4*)Wlds;
    for (int i = threadIdx.x; i < TOT4; i += 256) Lv[i] = Wv[i];
  }
  __syncthreads();

  int wave = threadIdx.x >> 5;
  int lane = threadIdx.x & 31;
  int r0 = (blockIdx.x * 8 + wave) * 16;
  if (r0 >= N) return;                      // wave-uniform exit (after barrier)
  int subl = lane & 15;
  int hi = lane >> 4;                       // 0: K {k0,k0+1}, rows M..M+7 ; 1: K {k0+2,k0+3}, rows M+8..M+15
  int arow = r0 + subl;
  if (arow >= N) arow = N - 1;              // clamp: unconditional loads, store-side guard

  // A fragments for the whole K=64 sweep: 16 x v2f (b64 loads), reused by every N-tile
  v2f afrag[16];
  const float* Xr = X + (size_t)arow * 64 + 2 * hi;
#pragma unroll
  for (int kk = 0; kk < 16; ++kk) afrag[kk] = *(const v2f*)(Xr + 4 * kk);

#pragma unroll
  for (int n0 = 0; n0 < OUTC; n0 += 16) {
    v8f c = {};
#pragma unroll
    for (int kk = 0; kk < 16; ++kk) {
      int ka = 4 * kk + 2 * hi;
      v2f bb;
      bb.x = Wlds[ka * OUTC + n0 + subl];          // ds_load_b32, immediate offsets
      bb.y = Wlds[(ka + 1) * OUTC + n0 + subl];
      c = __builtin_amdgcn_wmma_f32_16x16x4_f32(false, afrag[kk], false, bb, (short)0, c, false, false);
    }
    int col = n0 + subl;
    float bj = bias[col];
#pragma unroll
    for (int r = 0; r < 8; ++r) {            // D: vgpr r -> row r0+r (+8 for hi lanes)
      int row = r0 + r + 8 * hi;
      if (row < N) Y[(size_t)row * OUTC + col] = c[r] + bj;
    }
  }
}

// ---------------- edge kernels ----------------

__device__ inline void atomicMaxFloat(float* addr, float val) {
  // sign-aware monotonic punning; buffer initialized to -inf
  if (val >= 0.0f) atomicMax((int*)addr, __float_as_int(val));
  else             atomicMin((unsigned int*)addr, __float_as_uint(val));
}

// One thread per (edge, head): e = sum_c att[h,c]*leakyrelu(xl[src]+xr[dst]); scatter-max
__global__ void edge_e_kernel(const float* __restrict__ xl, const float* __restrict__ xr,
                              const int* __restrict__ src, const int* __restrict__ dst,
                              const float* __restrict__ att, float* __restrict__ e,
                              float* __restrict__ emax, int E, int H, int C) {
  int idx = blockIdx.x * blockDim.x + threadIdx.x;
  if (idx >= E * H) return;
  int ed = idx / H, h = idx - ed * H;
  int s = src[ed], d = dst[ed];
  const float4* ps = (const float4*)(xl + (size_t)s * (H * C) + h * C);
  const float4* pd = (const float4*)(xr + (size_t)d * (H * C) + h * C);
  const float4* pa = (const float4*)(att + h * C);
  float a = 0.0f;
  for (int c4 = 0; c4 < C / 4; ++c4) {
    float4 u = ps[c4], v = pd[c4], w = pa[c4];
    float t;
    t = u.x + v.x; t = t > 0.0f ? t : 0.2f * t; a = fmaf(w.x, t, a);
    t = u.y + v.y; t = t > 0.0f ? t : 0.2f * t; a = fmaf(w.y, t, a);
    t = u.z + v.z; t = t > 0.0f ? t : 0.2f * t; a = fmaf(w.z, t, a);
    t = u.w + v.w; t = t > 0.0f ? t : 0.2f * t; a = fmaf(w.w, t, a);
  }
  e[idx] = a;
  atomicMaxFloat(&emax[(size_t)d * H + h], a);
}

// e <- exp(e - emax[dst]); scatter-add into denom[Nd,H]
__global__ void edge_exp_kernel(float* __restrict__ e, const int* __restrict__ dst,
                                const float* __restrict__ emax, float* __restrict__ denom,
                                int E, int H) {
  int idx = blockIdx.x * blockDim.x + threadIdx.x;
  if (idx >= E * H) return;
  int ed = idx / H, h = idx - ed * H;
  int d = dst[ed];
  float ex = __expf(e[idx] - emax[(size_t)d * H + h]);
  e[idx] = ex;
  atomicAdd(&denom[(size_t)d * H + h], ex);
}

// acc[dst, h*C+c] += (e/denom) * xl[src, h*C+c]  (one thread per (edge, head))
__global__ void edge_acc_kernel(const float* __restrict__ e, const float* __restrict__ xl,
                                const int* __restrict__ src, const int* __restrict__ dst,
                                const float* __restrict__ denom, float* __restrict__ acc,
                                int E, int H, int C) {
  int idx = blockIdx.x * blockDim.x + threadIdx.x;
  if (idx >= E * H) return;
  int ed = idx / H, h = idx - ed * H;
  int s = src[ed], d = dst[ed];
  float alpha = e[idx] / (denom[(size_t)d * H + h] + 1e-16f);
  const float4* xs = (const float4*)(xl + (size_t)s * (H * C) + h * C);
  float* ad = acc + (size_t)d * (H * C) + h * C;
  for (int c4 = 0; c4 < C / 4; ++c4) {
    float4 u = xs[c4];
    atomicAdd(&ad[4 * c4 + 0], alpha * u.x);
    atomicAdd(&ad[4 * c4 + 1], alpha * u.y);
    atomicAdd(&ad[4 * c4 + 2], alpha * u.z);
    atomicAdd(&ad[4 * c4 + 3], alpha * u.w);
  }
}

// out = ((acc + b0 + b1? + b2?) * invT), optional ELU
__global__ void finalize_kernel(const float* __restrict__ acc, const float* __restrict__ b0,
                                const float* b1, const float* b2, float invT,
                                float* __restrict__ out, int N, int W, int do_elu) {
  int idx = blockIdx.x * blockDim.x + threadIdx.x;
  if (idx >= N * W) return;
  int j = idx % W;
  float s = acc[idx] + b0[j];
  if (b1) s += b1[j];
  if (b2) s += b2[j];
  s *= invT;
  if (do_elu) s = s > 0.0f ? s : expm1f(s);
  out[idx] = s;
}

// ---------------- host orchestration ----------------

extern "C" void kernel_launch(void* const* d_in, const int* in_sizes, int n_in,
                              void* d_out, int out_size, void* d_ws, size_t ws_size,
                              hipStream_t stream) {
  (void)n_in; (void)out_size; (void)ws_size;
  const int NV = 100000, NG = 20000, NC = 5000, NP = 10000;
  const int OFF_V = 0, OFF_G = NV, OFF_C = NV + NG, OFF_P = NV + NG + NC;
  const int NTOT = NV + NG + NC + NP;

  float* ws   = (float*)d_ws;
  float* h    = ws;                              // [NTOT,64]
  float* h2   = h    + (size_t)NTOT * 64;        // [NTOT,64]
  float* acc  = h2   + (size_t)NTOT * 64;        // [NTOT,64] (layer2 uses width 32)
  float* xl   = acc  + (size_t)NTOT * 64;        // [<=NV,64]
  float* xr   = xl   + (size_t)NV * 64;          // [<=NV,64]
  float* ebuf = xr   + (size_t)NV * 64;          // [<=500000,4]
  float* emax = ebuf + (size_t)500000 * 4;       // [<=NV*4]
  float* denom= emax + (size_t)NV * 4;           // [<=NV*4]

  auto F = [&](int i) { return (const float*)d_in[i]; };
  auto cdiv = [](long a, long b) { return (unsigned)((a + b - 1) / b); };

  // params flattened by sorted dict keys: conv1(10..45), conv2(46..81), proj(82..89)
  // edge order within conv*: cg(0) gc(1) gp(2) gv(3) pg(4) vg(5)
  // field order: att(0) b_dst(1) b_src(2) bias(3) w_dst(4) w_src(5)
  const int P1 = 10, P2 = 46, PP = 82;

  // ---- projections to 64-dim ----
  proj_kernel<<<cdiv((long)NV * 64, TPB), TPB, 0, stream>>>(F(0), F(PP + 7), F(PP + 6), h + (size_t)OFF_V * 64, NV, 10);
  proj_kernel<<<cdiv((long)NG * 64, TPB), TPB, 0, stream>>>(F(1), F(PP + 3), F(PP + 2), h + (size_t)OFF_G * 64, NG, 3);
  proj_kernel<<<cdiv((long)NC * 64, TPB), TPB, 0, stream>>>(F(2), F(PP + 1), F(PP + 0), h + (size_t)OFF_C * 64, NC, 4);
  proj_kernel<<<cdiv((long)NP * 64, TPB), TPB, 0, stream>>>(F(3), F(PP + 5), F(PP + 4), h + (size_t)OFF_P * 64, NP, 64);

  struct ETm { int ei, pp, soff, Ns, doff, Nd; };
  const ETm ets[6] = {
    {4, 5, OFF_V, NV, OFF_G, NG},   // vg
    {5, 3, OFF_G, NG, OFF_V, NV},   // gv
    {6, 1, OFF_G, NG, OFF_C, NC},   // gc
    {7, 0, OFF_C, NC, OFF_G, NG},   // cg
    {8, 2, OFF_G, NG, OFF_P, NP},   // gp
    {9, 4, OFF_P, NP, OFF_G, NG},   // pg
  };

  auto run_layer = [&](const float* hin, int pbase, int H, int C) {
    int Wd = H * C;
    fill_kernel<<<cdiv((long)NTOT * Wd, TPB), TPB, 0, stream>>>(acc, 0.0f, NTOT * Wd);
    for (int t = 0; t < 6; ++t) {
      const ETm& et = ets[t];
      int pb = pbase + 6 * et.pp;
      int E = in_sizes[et.ei] / 2;
      const int* src = (const int*)d_in[et.ei];
      const int* dst = src + E;
      if (Wd == 64) {
        gemm_bias_wmma<64><<<cdiv(et.Ns, 128), TPB, 0, stream>>>(hin + (size_t)et.soff * 64, F(pb + 5), F(pb + 2), xl, et.Ns);
        gemm_bias_wmma<64><<<cdiv(et.Nd, 128), TPB, 0, stream>>>(hin + (size_t)et.doff * 64, F(pb + 4), F(pb + 1), xr, et.Nd);
      } else {
        gemm_bias_wmma<32><<<cdiv(et.Ns, 128), TPB, 0, stream>>>(hin + (size_t)et.soff * 64, F(pb + 5), F(pb + 2), xl, et.Ns);
        gemm_bias_wmma<32><<<cdiv(et.Nd, 128), TPB, 0, stream>>>(hin + (size_t)et.doff * 64, F(pb + 4), F(pb + 1), xr, et.Nd);
      }
      fill_kernel<<<cdiv((long)et.Nd * H, TPB), TPB, 0, stream>>>(emax, -INFINITY, et.Nd * H);
      fill_kernel<<<cdiv((long)et.Nd * H, TPB), TPB, 0, stream>>>(denom, 0.0f, et.Nd * H);
      edge_e_kernel<<<cdiv((long)E * H, TPB), TPB, 0, stream>>>(xl, xr, src, dst, F(pb + 0), ebuf, emax, E, H, C);
      edge_exp_kernel<<<cdiv((long)E * H, TPB), TPB, 0, stream>>>(ebuf, dst, emax, denom, E, H);
      edge_acc_kernel<<<cdiv((long)E * H, TPB), TPB, 0, stream>>>(ebuf, xl, src, dst, denom,
                                                                  acc + (size_t)et.doff * Wd, E, H, C);
    }
  };

  // ---- conv1 (H=4, C=16, concat -> 64) + hetero mean + ELU ----
  run_layer(h, P1, 4, 16);
  finalize_kernel<<<cdiv((long)NV * 64, TPB), TPB, 0, stream>>>(acc + (size_t)OFF_V * 64, F(P1 + 6*3 + 3), nullptr, nullptr, 1.0f,       h2 + (size_t)OFF_V * 64, NV, 64, 1); // variant <- gv
  finalize_kernel<<<cdiv((long)NG * 64, TPB), TPB, 0, stream>>>(acc + (size_t)OFF_G * 64, F(P1 + 6*5 + 3), F(P1 + 6*0 + 3), F(P1 + 6*4 + 3), 1.0f/3.0f, h2 + (size_t)OFF_G * 64, NG, 64, 1); // gene <- vg,cg,pg
  finalize_kernel<<<cdiv((long)NC * 64, TPB), TPB, 0, stream>>>(acc + (size_t)OFF_C * 64, F(P1 + 6*1 + 3), nullptr, nullptr, 1.0f,       h2 + (size_t)OFF_C * 64, NC, 64, 1); // complex <- gc
  finalize_kernel<<<cdiv((long)NP * 64, TPB), TPB, 0, stream>>>(acc + (size_t)OFF_P * 64, F(P1 + 6*2 + 3), nullptr, nullptr, 1.0f,       h2 + (size_t)OFF_P * 64, NP, 64, 1); // phenotype <- gp

  // ---- conv2 (H=1, C=32) + hetero mean -> output ----
  run_layer(h2, P2, 1, 32);
  float* out = (float*)d_out;
  finalize_kernel<<<cdiv((long)NV * 32, TPB), TPB, 0, stream>>>(acc + (size_t)OFF_V * 32, F(P2 + 6*3 + 3), nullptr, nullptr, 1.0f,       out,                               NV, 32, 0);
  finalize_kernel<<<cdiv((long)NG * 32, TPB), TPB, 0, stream>>>(acc + (size_t)OFF_G * 32, F(P2 + 6*5 + 3), F(P2 + 6*0 + 3), F(P2 + 6*4 + 3), 1.0f/3.0f, out + (size_t)NV * 32,             NG, 32, 0);
  finalize_kernel<<<cdiv((long)NC * 32, TPB), TPB, 0, stream>>>(acc + (size_t)OFF_C * 32, F(P2 + 6*1 + 3), nullptr, nullptr, 1.0f,       out + (size_t)(NV + NG) * 32,      NC, 32, 0);
  finalize_kernel<<<cdiv((long)NP * 32, TPB), TPB, 0, stream>>>(acc + (size_t)OFF_P * 32, F(P2 + 6*2 + 3), nullptr, nullptr, 1.0f,       out + (size_t)(NV + NG + NC) * 32, NP, 32, 0);
}